// AttentionMM_5179730559505
// MI455X (gfx1250) — compile-verified
//
#include <hip/hip_runtime.h>
#include <math.h>

// B=64, T=512, E=512; inputs: m1(B,T,E), m2(B,T,E), W1(B,E,1), b1(B,T), W2(B,E,1), b2(B,T)
// output: concat(v1, v2) -> (B, 2E) float32

typedef __attribute__((ext_vector_type(2))) float v2f;
typedef __attribute__((ext_vector_type(8))) float v8f;

#define TT 512
#define EE 512

__device__ __forceinline__ float wave_reduce_add(float v) {
#pragma unroll
  for (int off = 16; off > 0; off >>= 1) v += __shfl_xor(v, off, 32);
  return v;
}
__device__ __forceinline__ float wave_reduce_max(float v) {
#pragma unroll
  for (int off = 16; off > 0; off >>= 1) v = fmaxf(v, __shfl_xor(v, off, 32));
  return v;
}

__global__ __launch_bounds__(512)
void attn_mm_fused_kernel(const float* __restrict__ m1, const float* __restrict__ m2,
                          const float* __restrict__ W1, const float* __restrict__ b1,
                          const float* __restrict__ W2, const float* __restrict__ b2,
                          float* __restrict__ out) {
  const int b    = blockIdx.x;
  const int tid  = threadIdx.x;     // 0..511
  const int wid  = tid >> 5;        // 0..15 (wave32)
  const int lane = tid & 31;

  const float* m1b = m1 + (size_t)b * TT * EE;
  const float* m2b = m2 + (size_t)b * TT * EE;

  __shared__ __align__(16) float u1[TT];
  __shared__ __align__(16) float u2[TT];
  __shared__ __align__(16) float w1[EE];
  __shared__ __align__(16) float w2[EE];
  __shared__ __align__(16) float s1[TT];
  __shared__ __align__(16) float s2[TT];
  __shared__ float red[16];

  // ---------- Phase A: u1[t] = tanh(m1[t,:].W1 + b1[t]), u2 likewise ----------
  // waves 0..7 -> m1/u1 ; waves 8..15 -> m2/u2 ; each wave does 64 rows.
  {
    const bool  first = (wid < 8);
    const float* Mb   = first ? m1b : m2b;
    const float* Wb   = (first ? W1 : W2) + (size_t)b * EE;
    const float* Bb   = (first ? b1 : b2) + (size_t)b * TT;
    float*       ul   = first ? u1 : u2;
    const int    wrel = wid & 7;
    float4 wf[4];
#pragma unroll
    for (int i = 0; i < 4; ++i)
      wf[i] = *(const float4*)(Wb + lane * 4 + i * 128);
    for (int r = 0; r < 64; ++r) {
      const int t = wrel * 64 + r;
      const float* row = Mb + t * EE;
      float acc = 0.f;
#pragma unroll
      for (int i = 0; i < 4; ++i) {
        float4 mv = *(const float4*)(row + lane * 4 + i * 128);
        acc += mv.x * wf[i].x + mv.y * wf[i].y + mv.z * wf[i].z + mv.w * wf[i].w;
      }
      acc = wave_reduce_add(acc);
      if (lane == 0) ul[t] = tanhf(acc + Bb[t]);
    }
  }
  __syncthreads();

  // ---------- Phase B: w1[e] = sum_t m1[t,e]*u1[t]; w2[e] = sum_t m2[t,e]*u2[t] ----------
  {
    float a1 = 0.f, a2 = 0.f;
#pragma unroll 8
    for (int t = 0; t < TT; ++t) {
      a1 += m1b[t * EE + tid] * u1[t];
      a2 += m2b[t * EE + tid] * u2[t];
    }
    w1[tid] = a1;
    w2[tid] = a2;
  }
  __syncthreads();

  // ---------- Phase C (WMMA f32 16x16x4): scores1 = m1 . w2 ; scores2 = m2 . w1 ----------
  // 64 tiles of 16 rows (32 for each matrix); each wave takes tiles wid, wid+16, wid+32, wid+48.
  // B-matrix columns are all identical (= the gating vector), so D columns are identical dots.
  {
    for (int ttile = wid; ttile < 64; ttile += 16) {
      const bool   first = (ttile < 32);
      const float* M     = first ? m1b : m2b;
      const float* wv    = first ? w2 : w1;
      float*       sc    = first ? s1 : s2;
      const int    row0  = (ttile & 31) * 16;
      const int    half  = lane >> 4;   // 0: K0/K1 pair, 1: K2/K3 pair
      const int    mrow  = lane & 15;   // M index within tile
      const float* rowp  = M + (row0 + mrow) * EE + 2 * half;
      const float* wvp   = wv + 2 * half;
      v8f c = {};
      for (int k = 0; k < EE; k += 4) {
        v2f a = *(const v2f*)(rowp + k);           // A fragment: 16x4 f32 tile
        v2f bb;
        bb.x = wvp[k];                             // B row-k value (same for all N)
        bb.y = wvp[k + 1];
        c = __builtin_amdgcn_wmma_f32_16x16x4_f32(
            /*neg_a=*/false, a, /*neg_b=*/false, bb,
            /*c_mod=*/(short)0, c, /*reuse_a=*/false, /*reuse_b=*/false);
      }
      if (mrow == 0) {                             // lane 0: M=0..7, lane 16: M=8..15
#pragma unroll
        for (int r = 0; r < 8; ++r) sc[row0 + half * 8 + r] = c[r];
      }
    }
  }
  __syncthreads();

  // ---------- Phase D: softmax over t for s1 and s2 ----------
  {
    float* sarr0 = s1;
    float* sarr1 = s2;
#pragma unroll
    for (int which = 0; which < 2; ++which) {
      float* s = (which == 0) ? sarr0 : sarr1;
      float v = s[tid];
      float m = wave_reduce_max(v);
      if (lane == 0) red[wid] = m;
      __syncthreads();
      if (tid == 0) {
        float mm = red[0];
        for (int i = 1; i < 16; ++i) mm = fmaxf(mm, red[i]);
        red[0] = mm;
      }
      __syncthreads();
      const float gmax = red[0];
      __syncthreads();
      float e = __expf(v - gmax);
      float ps = wave_reduce_add(e);
      if (lane == 0) red[wid] = ps;
      __syncthreads();
      if (tid == 0) {
        float ss = 0.f;
        for (int i = 0; i < 16; ++i) ss += red[i];
        red[0] = ss;
      }
      __syncthreads();
      s[tid] = e / red[0];
      __syncthreads();
    }
  }

  // ---------- Phase E: v1[e] = sum_t m1[t,e]*alpha1[t]; v2 likewise; concat out ----------
  {
    float a1 = 0.f, a2 = 0.f;
#pragma unroll 8
    for (int t = 0; t < TT; ++t) {
      a1 += m1b[t * EE + tid] * s1[t];
      a2 += m2b[t * EE + tid] * s2[t];
    }
    float* ob = out + (size_t)b * (2 * EE);
    ob[tid]      = a1;
    ob[EE + tid] = a2;
  }
}

extern "C" void kernel_launch(void* const* d_in, const int* in_sizes, int n_in,
                              void* d_out, int out_size, void* d_ws, size_t ws_size,
                              hipStream_t stream) {
  const float* m1 = (const float*)d_in[0];
  const float* m2 = (const float*)d_in[1];
  const float* W1 = (const float*)d_in[2];
  const float* b1 = (const float*)d_in[3];
  const float* W2 = (const float*)d_in[4];
  const float* b2 = (const float*)d_in[5];
  float* out = (float*)d_out;

  attn_mm_fused_kernel<<<dim3(64), dim3(512), 0, stream>>>(m1, m2, W1, b1, W2, b2, out);
}